// DPSOG_dimer_47579647705262
// MI455X (gfx1250) — compile-verified
//
#include <hip/hip_runtime.h>
#include <hip/hip_bf16.h>

// DeePMD-like energy+forces on MI455X (gfx1250, wave32, WMMA).
// All per-atom GEMMs (Q-net fwd/bwd, fit layer-1) run through
// V_WMMA_F32_16X16X4_F32 with LDS-staged, fully padded (branchless) tiles.

#define BB    4
#define NA    2048
#define KN    32
#define ATOMS (BB*NA)          // 8192 rows for all GEMMs (multiple of 16)
#define LBOX  40.0f
#define PIF   3.14159265358979323846f

typedef float v2f __attribute__((ext_vector_type(2)));
typedef float v8f __attribute__((ext_vector_type(8)));

struct NetPtrs {
  const float* eW[5]; const float* eB[5];   // emb
  const float* dW[5]; const float* dB[5];   // emb_dir
  const float* fW[5]; const float* fB[5];   // fit
  const float* linW;  const float* linB;
  const float* qW[4]; const float* qB[4];   // Q net
  const float* sogS;  const float* sogW;
};

// ---------------- tiny pyramid-net helpers (fully unrolled, registers) -----
template<int DIN>
__device__ __forceinline__ void pyr_fwd(const float* __restrict__ W, const float* __restrict__ Bv,
                                        const float* x, float* xo) {
  constexpr int DOUT = 2*DIN;
  #pragma unroll
  for (int c=0;c<DOUT;++c){
    float z = Bv[c];
    #pragma unroll
    for (int j=0;j<DIN;++j) z += x[j]*W[j*DOUT+c];
    xo[c] = x[c & (DIN-1)] + tanhf(z);     // concat-skip: dout == 2*din
  }
}

template<int DIN>
__device__ __forceinline__ void pyr_bwd(const float* __restrict__ W,
                                        const float* xin, const float* xout,
                                        const float* gout, float* gin) {
  constexpr int DOUT = 2*DIN;
  float gz[DOUT];
  #pragma unroll
  for (int c=0;c<DOUT;++c){
    float y = xout[c] - xin[c & (DIN-1)];  // recover tanh output
    gz[c] = gout[c]*(1.f - y*y);
  }
  #pragma unroll
  for (int j=0;j<DIN;++j){
    float g = gout[j] + gout[j+DIN];       // adjoint of concat([x,x])
    #pragma unroll
    for (int c=0;c<DOUT;++c) g += gz[c]*W[j*DOUT+c];
    gin[j] = g;
  }
}

__device__ __forceinline__ void emb_fwd_all(const float* const* W, const float* const* Bv, float x0,
                                            float* x1, float* x2, float* x3, float* x4, float* x5){
  float x0a[1] = {x0};
  pyr_fwd<1 >(W[0],Bv[0],x0a,x1);
  pyr_fwd<2 >(W[1],Bv[1],x1 ,x2);
  pyr_fwd<4 >(W[2],Bv[2],x2 ,x3);
  pyr_fwd<8 >(W[3],Bv[3],x3 ,x4);
  pyr_fwd<16>(W[4],Bv[4],x4 ,x5);
}

__device__ __forceinline__ float emb_bwd_all(const float* const* W, float x0,
                                             const float* x1,const float* x2,const float* x3,
                                             const float* x4,const float* x5,const float* g5){
  float g4[16], g3[8], g2[4], g1[2], g0[1];
  pyr_bwd<16>(W[4], x4, x5, g5, g4);
  pyr_bwd<8 >(W[3], x3, x4, g4, g3);
  pyr_bwd<4 >(W[2], x2, x3, g3, g2);
  pyr_bwd<2 >(W[1], x1, x2, g2, g1);
  float x0a[1] = {x0};
  pyr_bwd<1 >(W[0], x0a, x1, g1, g0);
  return g0[0];
}

template<int DIN,int DOUT>
__device__ __forceinline__ void dense_tanh(const float* __restrict__ W, const float* __restrict__ Bv,
                                           const float* x, float* y){
  #pragma unroll
  for (int c=0;c<DOUT;++c){
    float z = Bv[c];
    #pragma unroll
    for (int j=0;j<DIN;++j) z += x[j]*W[j*DOUT+c];
    y[c] = tanhf(z);
  }
}

// ---------------- weight packing: zero-padded (optionally transposed) ------
__global__ void k_pack_w(const float* __restrict__ W, int rows, int cols, int trans,
                         float* __restrict__ out, int Kp, int Np){
  int t = blockIdx.x*blockDim.x + threadIdx.x;
  if (t >= Kp*Np) return;
  int k = t / Np, n = t % Np;
  float v = 0.f;
  if (!trans){ if (k < rows && n < cols) v = W[(size_t)k*cols + n]; }  // out = W (KxN)
  else       { if (k < cols && n < rows) v = W[(size_t)n*cols + k]; }  // out = W^T
  out[t] = v;
}

__global__ void k_pack_b(const float* __restrict__ b, int n, float* __restrict__ out, int np){
  int t = blockIdx.x*blockDim.x + threadIdx.x;
  if (t < np) out[t] = (t < n) ? b[t] : 0.f;
}

// ---------------- WMMA GEMM (padded, branchless, LDS-staged) ---------------
// Y[ATOMS x Np] = act(X[ATOMS x Kp] * Wp[Kp x Np] + bias). One wave / 16x16 tile.
// A frag: lane l holds A[m=l%16][k=2*(l/16)+v]; B frag: B[k=2*(l/16)+v][n=l%16].
template<bool TANH, bool GATE>
__global__ __launch_bounds__(32)
void k_wmma_gemm(const float* __restrict__ X, int lda,
                 const float* __restrict__ Wp, int ldw,
                 const float* __restrict__ bias,
                 const float* __restrict__ gate,
                 float* __restrict__ Y, int ldc, int Kp){
  __shared__ float sX[16][16];   // m-major X tile
  __shared__ float sB[16][16];   // k-major W tile
  const int lane = threadIdx.x;
  const int m0 = blockIdx.x*16;
  const int n0 = blockIdx.y*16;
  const int mr = lane & 15;
  const int kh = lane >> 4;
  const float bn = bias ? bias[n0+mr] : 0.f;
  v8f acc = {};
  for (int kb = 0; kb < Kp; kb += 16){
    #pragma unroll
    for (int i=0;i<2;++i){                       // 2 x b128 per lane per tile
      int t  = i*32 + lane;
      int rw = t >> 2;
      int c4 = (t & 3) << 2;
      *(float4*)&sX[rw][c4] = *(const float4*)(X  + (size_t)(m0+rw)*lda + kb + c4);
      *(float4*)&sB[rw][c4] = *(const float4*)(Wp + (size_t)(kb+rw)*ldw + n0 + c4);
    }
    __syncthreads();                             // 1-wave group: cheap ordering
    #pragma unroll
    for (int k2=0; k2<16; k2+=4){
      v2f a = *(const v2f*)&sX[mr][k2 + 2*kh];   // ds_load_b64
      v2f b;
      b[0] = sB[k2 + 2*kh + 0][mr];
      b[1] = sB[k2 + 2*kh + 1][mr];
      acc = __builtin_amdgcn_wmma_f32_16x16x4_f32(false, a, false, b, (short)0, acc, false, false);
    }
    __syncthreads();
  }
  #pragma unroll
  for (int rr=0; rr<8; ++rr){
    int m = m0 + rr + (kh<<3);
    int n = n0 + mr;
    float v = acc[rr] + bn;
    if (TANH) v = tanhf(v);
    if (GATE){ float g = gate[(size_t)m*ldc + n]; v *= (1.f - g*g); }
    Y[(size_t)m*ldc + n] = v;
  }
}

// ---------------- gen_coor forward: RIa (.,K,4) and SR (.,K) ---------------
__global__ void k_gencoor_fwd(const float* __restrict__ coords, const int* __restrict__ neigh,
                              const int* __restrict__ rad,
                              float* __restrict__ RIa, float* __restrict__ SR){
  int gid = blockIdx.x*blockDim.x + threadIdx.x;
  if (gid >= ATOMS*KN) return;
  int b = gid / (NA*KN);
  int r = gid % (NA*KN);
  int i = r / KN;
  int j = neigh[gid];
  int jj = (j < 0) ? 0 : j;
  float rc = (float)rad[0], rcs = 0.75f*rc;
  const float* cb = coords + (size_t)b*NA*3;
  float dx = cb[jj*3+0]-cb[i*3+0];
  float dy = cb[jj*3+1]-cb[i*3+1];
  float dz = cb[jj*3+2]-cb[i*3+2];
  dx -= LBOX*rintf(dx*(1.f/LBOX));
  dy -= LBOX*rintf(dy*(1.f/LBOX));
  dz -= LBOX*rintf(dz*(1.f/LBOX));
  float d2 = dx*dx+dy*dy+dz*dz;
  float d  = sqrtf(fmaxf(d2, 1e-12f));
  float invd = 1.f/d;
  float sw = 0.f;
  if (d < rcs) sw = invd;
  else if (d < rc){
    float tt = (d-rcs)/(rc-rcs);
    sw = (0.5f*cosf(PIF*tt)+0.5f)*invd;
  }
  if (j < 0) sw = 0.f;
  RIa[(size_t)gid*4+0] = sw;
  RIa[(size_t)gid*4+1] = sw*dx*invd;
  RIa[(size_t)gid*4+2] = sw*dy*invd;
  RIa[(size_t)gid*4+3] = sw*dz*invd;
  SR[gid] = sw;                          // av=0, std=1, mask folded in
}

// ---------------- per-atom embeddings + Gram descriptor (1 wave/atom) ------
__global__ void k_embed_fwd(NetPtrs np, const float* __restrict__ RIa, const float* __restrict__ SR,
                            float* __restrict__ Tb, float* __restrict__ Dcona){
  __shared__ float sG [KN][KN+1];
  __shared__ float sGi[KN][KN+1];
  __shared__ float sR [KN][4];
  __shared__ float sT [4][KN];
  int atom = blockIdx.x;
  int k = threadIdx.x;
  size_t idx = (size_t)atom*KN + k;
  float sr = SR[idx];
  float x1[2],x2[4],x3[8],x4[16],x5[32];
  emb_fwd_all(np.eW, np.eB, sr, x1,x2,x3,x4,x5);
  #pragma unroll
  for (int c=0;c<KN;++c) sG[k][c]=x5[c];
  emb_fwd_all(np.dW, np.dB, sr, x1,x2,x3,x4,x5);
  #pragma unroll
  for (int c=0;c<KN;++c) sGi[k][c]=x5[c];
  #pragma unroll
  for (int a=0;a<4;++a) sR[k][a]=RIa[idx*4+a];
  __syncthreads();
  int c = k;                                        // thread owns column c
  float Tc[4];
  #pragma unroll
  for (int a=0;a<4;++a){
    float acc=0.f;
    for (int kk=0;kk<KN;++kk) acc += sR[kk][a]*sG[kk][c];
    Tc[a]=acc; sT[a][c]=acc;
    Tb[(size_t)atom*128 + a*KN + c]=acc;            // save T for backward
  }
  float dr=0.f;
  for (int kk=0;kk<KN;++kk) dr += sGi[kk][c];
  dr *= (1.f/KN);
  __syncthreads();
  float* dco = Dcona + (size_t)atom*1056;
  for (int s=0;s<KN;++s){
    float acc=0.f;
    #pragma unroll
    for (int a=0;a<4;++a) acc += sT[a][s]*Tc[a];
    dco[s*KN + c] = acc*(1.f/(KN*KN));              // Da = T^T T / K^2
  }
  dco[1024 + c] = dr;                               // Dr appended
}

// ---------------- fit tail (16->8->4->2->1), per-atom energy ---------------
__global__ void k_fit_tail(NetPtrs np, const float* __restrict__ F1,
                           float* __restrict__ fita, float* __restrict__ Eout){
  int atom = blockIdx.x*blockDim.x + threadIdx.x;
  if (atom >= ATOMS) return;
  float y1[16];
  #pragma unroll
  for (int j=0;j<16;++j) y1[j]=F1[(size_t)atom*16+j];
  float y2[8], y3[4], y4[2], y5[1];
  dense_tanh<16,8>(np.fW[1],np.fB[1],y1,y2);
  dense_tanh<8 ,4>(np.fW[2],np.fB[2],y2,y3);
  dense_tanh<4 ,2>(np.fW[3],np.fB[3],y3,y4);
  dense_tanh<2 ,1>(np.fW[4],np.fB[4],y4,y5);
  float Ea = y5[0]*np.linW[0] + np.linB[0];
  atomicAdd(&Eout[atom/NA], Ea);
  float* fa = fita + (size_t)atom*16;
  #pragma unroll
  for (int j=0;j<8;++j) fa[j]=y2[j];
  #pragma unroll
  for (int j=0;j<4;++j) fa[8+j]=y3[j];
  fa[12]=y4[0]; fa[13]=y4[1]; fa[14]=y5[0]; fa[15]=0.f;
}

// ---------------- O(N^2) long-range: E_l, gQ, direct coord grads -----------
// Qv16/gQ16 are padded 16-col layouts (col 0 live) feeding branchless GEMMs.
__global__ void k_lr_pairs(NetPtrs np, const float* __restrict__ coords, const float* __restrict__ Qv16,
                           float* __restrict__ Eout, float* __restrict__ gQ16, float* __restrict__ gC){
  __shared__ float red[4][256];
  int b = blockIdx.x / NA;
  int i = blockIdx.x % NA;
  int tid = threadIdx.x;
  float wv[4], inv[4]; float k0=0.f;
  #pragma unroll
  for (int c=0;c<4;++c){ wv[c]=np.sogW[c]; float sg=np.sogS[c]; inv[c]=1.f/(2.f*sg*sg); k0+=wv[c]; }
  const float* cb = coords + (size_t)b*NA*3;
  const float* Qb = Qv16 + (size_t)b*NA*16;
  float cx=cb[i*3+0], cy=cb[i*3+1], cz=cb[i*3+2];
  float s=0.f, gx=0.f, gy=0.f, gz=0.f;
  for (int j=tid; j<NA; j+=256){
    if (j+256 < NA) __builtin_prefetch(&cb[(j+256)*3], 0, 0);
    float dx=cx-cb[j*3+0], dy=cy-cb[j*3+1], dz=cz-cb[j*3+2];
    dx -= LBOX*rintf(dx*(1.f/LBOX));
    dy -= LBOX*rintf(dy*(1.f/LBOX));
    dz -= LBOX*rintf(dz*(1.f/LBOX));
    float r2 = dx*dx+dy*dy+dz*dz;
    float kv=0.f, kp=0.f;
    #pragma unroll
    for (int c=0;c<4;++c){ float e=wv[c]*expf(-r2*inv[c]); kv+=e; kp-=inv[c]*e; }
    float qj = Qb[(size_t)j*16];
    s += kv*qj;
    float f = 2.f*qj*kp;
    gx += f*dx; gy += f*dy; gz += f*dz;
  }
  red[0][tid]=s; red[1][tid]=gx; red[2][tid]=gy; red[3][tid]=gz;
  __syncthreads();
  for (int st=128; st>0; st>>=1){
    if (tid<st){
      #pragma unroll
      for (int q=0;q<4;++q) red[q][tid]+=red[q][tid+st];
    }
    __syncthreads();
  }
  if (tid==0){
    float qi = Qb[(size_t)i*16];
    float S = red[0][0];
    atomicAdd(&Eout[b], 0.5f*(qi*S - k0*qi*qi));
    gQ16[((size_t)b*NA+i)*16] = S - k0*qi;
    float* g = gC + ((size_t)b*NA+i)*3;
    atomicAdd(&g[0], qi*red[1][0]);
    atomicAdd(&g[1], qi*red[2][0]);
    atomicAdd(&g[2], qi*red[3][0]);
  }
}

// ---------------- descriptor/embedding backward (1 wave/atom) --------------
__global__ void k_descr_bwd(NetPtrs np, const float* __restrict__ RIa, const float* __restrict__ SR,
                            const float* __restrict__ Tb, const float* __restrict__ F1,
                            const float* __restrict__ fita,
                            float* __restrict__ gRT, float* __restrict__ gSR){
  __shared__ float sDa[KN][KN+1];
  __shared__ float sDr[KN];
  __shared__ float sT [4][KN];
  __shared__ float sgT[4][KN];
  int atom = blockIdx.x;
  int t = threadIdx.x;
  // --- fit tail backward (tiny; redundant on all lanes) ---
  float y1[16];
  #pragma unroll
  for (int j=0;j<16;++j) y1[j]=F1[(size_t)atom*16+j];
  const float* fa = fita + (size_t)atom*16;
  float y2[8], y3[4], y4[2], y5;
  #pragma unroll
  for (int j=0;j<8;++j) y2[j]=fa[j];
  #pragma unroll
  for (int j=0;j<4;++j) y3[j]=fa[8+j];
  y4[0]=fa[12]; y4[1]=fa[13]; y5=fa[14];
  float gz5 = np.linW[0]*(1.f - y5*y5);
  float gz4[2];
  #pragma unroll
  for (int j=0;j<2;++j){ float g = gz5*np.fW[4][j]; gz4[j]=g*(1.f-y4[j]*y4[j]); }
  float gz3[4];
  #pragma unroll
  for (int j=0;j<4;++j){
    float acc=0.f;
    #pragma unroll
    for (int c=0;c<2;++c) acc += gz4[c]*np.fW[3][j*2+c];
    gz3[j]=acc*(1.f-y3[j]*y3[j]);
  }
  float gz2[8];
  #pragma unroll
  for (int j=0;j<8;++j){
    float acc=0.f;
    #pragma unroll
    for (int c=0;c<4;++c) acc += gz3[c]*np.fW[2][j*4+c];
    gz2[j]=acc*(1.f-y2[j]*y2[j]);
  }
  float gz1[16];
  #pragma unroll
  for (int j=0;j<16;++j){
    float acc=0.f;
    #pragma unroll
    for (int c=0;c<8;++c) acc += gz2[c]*np.fW[1][j*8+c];
    gz1[j]=acc*(1.f-y1[j]*y1[j]);
  }
  // --- gDcona column t (gz1 . W_fit1^T), split Da / Dr parts ---
  for (int s=0;s<KN;++s){
    const float* wr = np.fW[0] + (size_t)(s*KN+t)*16;
    float acc=0.f;
    #pragma unroll
    for (int l=0;l<16;++l) acc += gz1[l]*wr[l];
    sDa[s][t]=acc;
  }
  {
    const float* wr = np.fW[0] + (size_t)(1024+t)*16;
    float acc=0.f;
    #pragma unroll
    for (int l=0;l<16;++l) acc += gz1[l]*wr[l];
    sDr[t]=acc;
  }
  #pragma unroll
  for (int a=0;a<4;++a) sT[a][t] = Tb[(size_t)atom*128 + a*KN + t];
  __syncthreads();
  // --- gT = T (gDa + gDa^T) / K^2 ---
  #pragma unroll
  for (int a=0;a<4;++a){
    float acc=0.f;
    for (int s=0;s<KN;++s) acc += sT[a][s]*(sDa[s][t]+sDa[t][s]);
    sgT[a][t]=acc*(1.f/(KN*KN));
  }
  __syncthreads();
  // --- per-neighbor backward (k == t) ---
  size_t idx = (size_t)atom*KN + t;
  float sr = SR[idx];
  float ria[4];
  #pragma unroll
  for (int a=0;a<4;++a) ria[a]=RIa[idx*4+a];
  float x1[2],x2[4],x3[8],x4[16],x5[32];
  emb_fwd_all(np.eW,np.eB,sr,x1,x2,x3,x4,x5);      // recompute Gia
  float g5v[KN];
  for (int c=0;c<KN;++c){
    float gg=0.f;
    #pragma unroll
    for (int a=0;a<4;++a) gg += ria[a]*sgT[a][c];  // gGia = RIa^T gT
    g5v[c]=gg;
  }
  #pragma unroll
  for (int a=0;a<4;++a){
    float acc=0.f;
    for (int c=0;c<KN;++c) acc += sgT[a][c]*x5[c]; // gRIa(T path) = gT Gia^T
    gRT[idx*4+a]=acc;
  }
  float gs = emb_bwd_all(np.eW,sr,x1,x2,x3,x4,x5,g5v);
  emb_fwd_all(np.dW,np.dB,sr,x1,x2,x3,x4,x5);      // recompute Gi
  for (int c=0;c<KN;++c) g5v[c] = sDr[c]*(1.f/KN); // gGi = gDr / K
  gs += emb_bwd_all(np.dW,sr,x1,x2,x3,x4,x5,g5v);
  gSR[idx]=gs;
}

// ---------------- gen_coor backward: scatter coordinate grads --------------
__global__ void k_gencoor_bwd(const float* __restrict__ coords, const int* __restrict__ neigh,
                              const int* __restrict__ rad,
                              const float* __restrict__ gA0, const float* __restrict__ gRT,
                              const float* __restrict__ gSRb, float* __restrict__ gC){
  int gid = blockIdx.x*blockDim.x + threadIdx.x;
  if (gid >= ATOMS*KN) return;
  int b = gid / (NA*KN);
  int r = gid % (NA*KN);
  int i = r / KN;
  int k = r % KN;
  int j = neigh[gid];
  int jj = (j<0)?0:j;
  float rc=(float)rad[0], rcs=0.75f*rc;
  const float* cb = coords + (size_t)b*NA*3;
  float dx = cb[jj*3+0]-cb[i*3+0];
  float dy = cb[jj*3+1]-cb[i*3+1];
  float dz = cb[jj*3+2]-cb[i*3+2];
  dx -= LBOX*rintf(dx*(1.f/LBOX));
  dy -= LBOX*rintf(dy*(1.f/LBOX));
  dz -= LBOX*rintf(dz*(1.f/LBOX));
  float d2 = dx*dx+dy*dy+dz*dz;
  float d  = sqrtf(fmaxf(d2,1e-12f));
  float invd = 1.f/d;
  float sw=0.f, dsw=0.f;
  if (d < rcs){ sw=invd; dsw=-invd*invd; }
  else if (d < rc){
    float tt=(d-rcs)/(rc-rcs);
    float cc=0.5f*cosf(PIF*tt)+0.5f;
    sw = cc*invd;
    dsw = (-0.5f*PIF*sinf(PIF*tt)/(rc-rcs))*invd - cc*invd*invd;
  }
  if (j<0){ sw=0.f; dsw=0.f; }
  int atom = b*NA+i;
  float gv0 = gA0[(size_t)atom*128 + k*4 + 0] + gRT[(size_t)gid*4+0];
  float gv1 = gA0[(size_t)atom*128 + k*4 + 1] + gRT[(size_t)gid*4+1];
  float gv2 = gA0[(size_t)atom*128 + k*4 + 2] + gRT[(size_t)gid*4+2];
  float gv3 = gA0[(size_t)atom*128 + k*4 + 3] + gRT[(size_t)gid*4+3];
  float gs  = gSRb[gid];
  float ux=dx*invd, uy=dy*invd, uz=dz*invd;
  float gsw = gv0 + gs + gv1*ux + gv2*uy + gv3*uz;
  float gux = sw*gv1, guy = sw*gv2, guz = sw*gv3;
  float ud  = ux*gux + uy*guy + uz*guz;
  float gdx = gsw*dsw*ux + (gux - ux*ud)*invd;
  float gdy = gsw*dsw*uy + (guy - uy*ud)*invd;
  float gdz = gsw*dsw*uz + (guz - uz*ud)*invd;
  float* gj = gC + ((size_t)b*NA+jj)*3;
  float* gi = gC + ((size_t)atom)*3;
  if (j>=0){
    atomicAdd(&gj[0], gdx); atomicAdd(&gj[1], gdy); atomicAdd(&gj[2], gdz);
  }
  atomicAdd(&gi[0], -gdx); atomicAdd(&gi[1], -gdy); atomicAdd(&gi[2], -gdz);
}

__global__ void k_finalize(const float* __restrict__ gC, float* __restrict__ Fout){
  int t = blockIdx.x*blockDim.x + threadIdx.x;
  if (t < ATOMS*3) Fout[t] = -gC[t];
}

// ---------------- host side ------------------------------------------------
static inline void gemm(hipStream_t st, int mode,
                        const float* X, int lda, const float* Wp, int ldw,
                        const float* bias, const float* gate,
                        float* Y, int ldc, int Kp, int Np){
  dim3 g(ATOMS/16, Np/16), b(32);
  switch(mode){
    case 0: k_wmma_gemm<true ,false><<<g,b,0,st>>>(X,lda,Wp,ldw,bias,gate,Y,ldc,Kp); break; // tanh
    case 1: k_wmma_gemm<false,false><<<g,b,0,st>>>(X,lda,Wp,ldw,bias,gate,Y,ldc,Kp); break; // linear
    case 2: k_wmma_gemm<false,true ><<<g,b,0,st>>>(X,lda,Wp,ldw,bias,gate,Y,ldc,Kp); break; // gated bwd
  }
}

static inline void packw(hipStream_t st, const float* W, int rows, int cols, int trans,
                         float* out, int Kp, int Np){
  int n = Kp*Np;
  k_pack_w<<<(n+255)/256,256,0,st>>>(W,rows,cols,trans,out,Kp,Np);
}

extern "C" void kernel_launch(void* const* d_in, const int* in_sizes, int n_in,
                              void* d_out, int out_size, void* d_ws, size_t ws_size,
                              hipStream_t stream) {
  (void)out_size; (void)ws_size;
  auto F = [&](int i){ return (const float*)d_in[i]; };
  auto I = [&](int i){ return (const int*)d_in[i]; };
  // Identify flattening convention by size signature.
  int xi, ni, qi, ei, di, fi, li, si, ri;
  if (in_sizes[0] == BB*NA) {                  // jax pytree (sorted keys): charge_index first
    xi=1; ni=2; qi=3; ei=11; di=21; fi=31; li=41; si=43; ri=45;
  } else if (n_in > 1 && in_sizes[1] > 1000) { // coords first, params sorted (Q first)
    xi=0; qi=1; ei=9; di=19; fi=29; li=39; si=41; ni=44; ri=45;
  } else {                                     // full insertion order
    xi=0; ei=1; di=11; fi=21; li=31; qi=33; si=41; ni=44; ri=45;
  }
  NetPtrs np;
  for (int l=0;l<5;++l){
    np.eW[l]=F(ei+2*l); np.eB[l]=F(ei+2*l+1);
    np.dW[l]=F(di+2*l); np.dB[l]=F(di+2*l+1);
    np.fW[l]=F(fi+2*l); np.fB[l]=F(fi+2*l+1);
  }
  np.linW=F(li); np.linB=F(li+1);
  for (int l=0;l<4;++l){ np.qW[l]=F(qi+2*l); np.qB[l]=F(qi+2*l+1); }
  np.sogS=F(si); np.sogW=F(si+1);
  const float* coords = F(xi);
  const int*   neigh  = I(ni);
  const int*   rad    = I(ri);

  // workspace carve-up (floats; every chunk a multiple of 16 for b128 loads)
  float* w = (float*)d_ws;
  size_t off = 0;
  auto take = [&](size_t n){ float* p = w + off; off += n; return p; };
  float* RIa   = take((size_t)ATOMS*KN*4);   // doubles as Q-net input (8192x128)
  float* SR    = take((size_t)ATOMS*KN);
  float* Tb    = take((size_t)ATOMS*128);
  float* Dcona = take((size_t)ATOMS*1056);
  float* F1    = take((size_t)ATOMS*16);
  float* fita  = take((size_t)ATOMS*16);
  float* A1    = take((size_t)ATOMS*96);
  float* A2    = take((size_t)ATOMS*208);
  float* A3    = take((size_t)ATOMS*208);
  float* Qv16  = take((size_t)ATOMS*16);
  float* gQ16  = take((size_t)ATOMS*16);
  float* gA3   = take((size_t)ATOMS*208);
  float* gA2   = take((size_t)ATOMS*208);
  float* gA1   = take((size_t)ATOMS*96);
  float* gA0   = take((size_t)ATOMS*128);
  float* gRT   = take((size_t)ATOMS*KN*4);
  float* gSRb  = take((size_t)ATOMS*KN);
  float* gC    = take((size_t)ATOMS*3 + 16); // +pad to keep alignment
  // packed (zero-padded / transposed) weights + biases
  float* fW0p  = take(1056*16);
  float* qW0p  = take(128*96);
  float* qW1p  = take((size_t)96*208);
  float* qW2p  = take((size_t)208*208);
  float* qW3p  = take((size_t)208*16);
  float* qW3t  = take((size_t)16*208);
  float* qW2t  = take((size_t)208*208);
  float* qW1t  = take((size_t)208*96);
  float* qW0t  = take((size_t)96*128);
  float* qB1p  = take(208);
  float* qB2p  = take(208);
  float* qB3p  = take(16);

  float* Eout = (float*)d_out;           // [BB] energies, then forces
  hipMemsetAsync(Eout, 0, BB*sizeof(float), stream);
  hipMemsetAsync(gC,   0, (size_t)ATOMS*3*sizeof(float), stream);
  hipMemsetAsync(gQ16, 0, (size_t)ATOMS*16*sizeof(float), stream);

  // pack weights (tiny, runs every call; deterministic)
  packw(stream, np.fW[0], 1056, 16, 0, fW0p, 1056, 16);
  packw(stream, np.qW[0], 128,  96, 0, qW0p, 128,  96);
  packw(stream, np.qW[1], 96,  200, 0, qW1p, 96,  208);
  packw(stream, np.qW[2], 200, 200, 0, qW2p, 208, 208);
  packw(stream, np.qW[3], 200,   1, 0, qW3p, 208,  16);
  packw(stream, np.qW[3], 200,   1, 1, qW3t, 16,  208);
  packw(stream, np.qW[2], 200, 200, 1, qW2t, 208, 208);
  packw(stream, np.qW[1], 96,  200, 1, qW1t, 208,  96);
  packw(stream, np.qW[0], 128,  96, 1, qW0t, 96,  128);
  k_pack_b<<<1,256,0,stream>>>(np.qB[1], 200, qB1p, 208);
  k_pack_b<<<1,256,0,stream>>>(np.qB[2], 200, qB2p, 208);
  k_pack_b<<<1,256,0,stream>>>(np.qB[3], 1,   qB3p, 16);

  int tot = ATOMS*KN;
  k_gencoor_fwd<<<(tot+255)/256,256,0,stream>>>(coords, neigh, rad, RIa, SR);
  k_embed_fwd  <<<ATOMS,32,0,stream>>>(np, RIa, SR, Tb, Dcona);

  // fit layer-1 (K=1056 -> 16) via WMMA, then tiny tail + Ea
  gemm(stream,0, Dcona,1056, fW0p,16,  np.fB[0], nullptr, F1,16, 1056, 16);
  k_fit_tail<<<(ATOMS+255)/256,256,0,stream>>>(np, F1, fita, Eout);

  // Q-net forward (all WMMA; padded widths, branchless)
  gemm(stream,0, RIa,128, qW0p,96,   np.qB[0], nullptr, A1,96,   128, 96);
  gemm(stream,0, A1 ,96 , qW1p,208,  qB1p,     nullptr, A2,208,  96,  208);
  gemm(stream,0, A2 ,208, qW2p,208,  qB2p,     nullptr, A3,208,  208, 208);
  gemm(stream,1, A3 ,208, qW3p,16,   qB3p,     nullptr, Qv16,16, 208, 16);

  // long-range energy + gQ + direct coord grads
  k_lr_pairs<<<ATOMS,256,0,stream>>>(np, coords, Qv16, Eout, gQ16, gC);

  // Q-net backward (WMMA, pre-transposed padded weights, tanh' gating fused)
  gemm(stream,2, gQ16,16, qW3t,208, nullptr, A3, gA3,208, 16,  208);
  gemm(stream,2, gA3,208, qW2t,208, nullptr, A2, gA2,208, 208, 208);
  gemm(stream,2, gA2,208, qW1t,96,  nullptr, A1, gA1,96,  208, 96);
  gemm(stream,1, gA1,96 , qW0t,128, nullptr, nullptr, gA0,128, 96, 128);

  // descriptor / embedding backward, then scatter coordinate grads
  k_descr_bwd  <<<ATOMS,32,0,stream>>>(np, RIa, SR, Tb, F1, fita, gRT, gSRb);
  k_gencoor_bwd<<<(tot+255)/256,256,0,stream>>>(coords, neigh, rad, gA0, gRT, gSRb, gC);
  k_finalize   <<<((ATOMS*3)+255)/256,256,0,stream>>>(gC, Eout + BB);
}